// GraphEmbed_27582279975440
// MI455X (gfx1250) — compile-verified
//
#include <hip/hip_runtime.h>
#include <hip/hip_bf16.h>

typedef __attribute__((ext_vector_type(2))) float v2f;
typedef __attribute__((ext_vector_type(8))) float v8f;

// ---------------------------------------------------------------------------
// Kernel 1: zero the counter region of the workspace (colsum + degrees)
// ---------------------------------------------------------------------------
__global__ void k_zero(unsigned* __restrict__ p, long long n) {
    long long i = (long long)blockIdx.x * blockDim.x + threadIdx.x;
    long long stride = (long long)gridDim.x * blockDim.x;
    for (; i < n; i += stride) p[i] = 0u;
}

// ---------------------------------------------------------------------------
// Kernel 2: out/in degree histograms (u32 atomics, counters are L2 resident)
// ---------------------------------------------------------------------------
__global__ void k_degrees(const int* __restrict__ src, const int* __restrict__ dst,
                          unsigned* __restrict__ odeg, unsigned* __restrict__ ideg,
                          long long E) {
    long long i = (long long)blockIdx.x * blockDim.x + threadIdx.x;
    long long stride = (long long)gridDim.x * blockDim.x;
    for (; i < E; i += stride) {
        atomicAdd(&odeg[src[i]], 1u);
        atomicAdd(&ideg[dst[i]], 1u);
    }
}

// ---------------------------------------------------------------------------
// Kernel 3: gating MLP via f32 WMMA 16x16x4, one wave per block.
//   A tile: 16x256 f32 -> LDS via GLOBAL_LOAD_ASYNC_TO_LDS_B128 (ASYNCcnt)
//           row stride 260 floats (260 % 64 == 4 -> conflict-free b64 reads)
//   B:      Wg (256x7) staged transposed into 16 zero-padded rows, stride 260
//           -> unconditional ds_load_b64 B fragments, no EXEC games
//   D:      16x16 f32; cols 0..6 -> sigmoid * out_deg^-1/2; col 7 slot gets
//           in_deg^-1/2 so the edge kernel reads one packed 32B row per edge.
// ---------------------------------------------------------------------------
__global__ __launch_bounds__(32)
void k_gate_wmma(const float* __restrict__ x, const float* __restrict__ Wg,
                 const float* __restrict__ bg,
                 const unsigned* __restrict__ odeg, const unsigned* __restrict__ ideg,
                 float* __restrict__ hsc, int N, int nTiles) {
    __shared__ __align__(16) float sWgT[16 * 260];  // [n][k] transposed, padded
    __shared__ __align__(16) float sA[16 * 260];    // [m][k], padded

    const int lane = threadIdx.x;                   // 0..31 (wave32)

    // Stage Wg transposed: sWgT[n*260 + k] = Wg[k][n] for n<7, zero for n>=7
    for (int i = lane; i < 16 * 256; i += 32) {
        int n = i >> 8, k = i & 255;
        sWgT[n * 260 + k] = (n < 7) ? Wg[k * 7 + n] : 0.0f;
    }
    __syncthreads();

    const int m  = lane & 15;          // A row in tile == B column index
    const int ko = (lane >> 4) * 2;    // K sub-offset for upper half-wave
    const long long total = (long long)N * 256;

    // Per-lane LDS fragment base addresses (constant across tiles)
    const float* pA = &sA[m * 260 + ko];
    const float* pB = &sWgT[m * 260 + ko];

    for (int tile = blockIdx.x; tile < nTiles; tile += gridDim.x) {
        const long long base = (long long)tile * 16 * 256;
        const bool full = (base + 4096) <= total;   // whole tile in range

        if (full) {
            // Async DMA of the contiguous 16KB tile straight into LDS.
            // 32 wave-wide b128 ops -> ASYNCcnt <= 32 (< 63 cap).
            const unsigned ldsbase = (unsigned)(size_t)(&sA[0]);
            for (int i = lane; i < 1024; i += 32) {
                int r = i >> 6;                      // row 0..15
                int c = (i & 63) << 2;               // col 0..252
                unsigned laddr = ldsbase + (unsigned)(r * 260 + c) * 4u;
                unsigned long long gaddr =
                    (unsigned long long)(size_t)(x + base + (long long)i * 4);
                asm volatile("global_load_async_to_lds_b128 %0, %1, off"
                             :: "v"(laddr), "v"(gaddr) : "memory");
            }
            asm volatile("s_wait_asynccnt 0x0" ::: "memory");
        } else {
            // Ragged last tile: zero-filled scalar path.
            for (int i = lane; i < 1024; i += 32) {
                long long gi = base + (long long)i * 4;
                float4 v;
                if (gi + 3 < total) v = *(const float4*)(x + gi);
                else                v = make_float4(0.f, 0.f, 0.f, 0.f);
                int r = i >> 6;
                int c = (i & 63) << 2;
                *(float4*)&sA[r * 260 + c] = v;
            }
        }
        __syncthreads();

        v8f acc = {0.f, 0.f, 0.f, 0.f, 0.f, 0.f, 0.f, 0.f};
        #pragma unroll
        for (int k0 = 0; k0 < 256; k0 += 4) {
            v2f a = *(const v2f*)(pA + k0);   // A[m][k0+ko .. +1]
            v2f b = *(const v2f*)(pB + k0);   // B[k0+ko .. +1][m] (transposed)
            acc = __builtin_amdgcn_wmma_f32_16x16x4_f32(
                false, a, false, b, (short)0, acc, false, false);
        }
        __syncthreads();  // all lanes done reading sA before next tile DMA

        // Writeout. D layout: VGPR v -> row (v + 8*(lane>=16)), col = lane&15.
        const int col   = m;
        const int rbase = (lane >> 4) * 8;
        const int r0    = tile * 16;
        if (col < 8) {
            const bool feat = (col < 7);
            const float bgv = feat ? bg[col] : 0.0f;
            const unsigned* __restrict__ dp = feat ? odeg : ideg;
            #pragma unroll
            for (int v = 0; v < 8; ++v) {
                int node = r0 + rbase + v;
                if (node < N) {
                    unsigned dg = dp[node];
                    float rs = dg ? __frsqrt_rn((float)dg) : 0.0f;
                    float g  = 1.0f / (1.0f + __expf(-(acc[v] + bgv)));
                    float val = feat ? rs * g : rs;
                    hsc[(long long)node * 8 + col] = val;
                }
            }
        }
    }
}

// ---------------------------------------------------------------------------
// Kernel 4: colsum[j] = sum_e norm_dst[dst[e]] * h_scaled[src[e]][j]
// (h_scaled is 3.2 MB -> L2 resident; only index reads hit HBM)
// ---------------------------------------------------------------------------
__global__ __launch_bounds__(256)
void k_edges(const int* __restrict__ src, const int* __restrict__ dst,
             const float* __restrict__ hsc, float* __restrict__ colsum,
             long long E) {
    float a0 = 0.f, a1 = 0.f, a2 = 0.f, a3 = 0.f, a4 = 0.f, a5 = 0.f, a6 = 0.f;
    long long i = (long long)blockIdx.x * blockDim.x + threadIdx.x;
    long long stride = (long long)gridDim.x * blockDim.x;
    for (; i < E; i += stride) {
        int s = src[i];
        int d = dst[i];
        float w = hsc[(long long)d * 8 + 7];                 // norm_dst[d]
        const float4 h0 = *(const float4*)(hsc + (long long)s * 8);
        const float4 h1 = *(const float4*)(hsc + (long long)s * 8 + 4);
        a0 = fmaf(w, h0.x, a0);
        a1 = fmaf(w, h0.y, a1);
        a2 = fmaf(w, h0.z, a2);
        a3 = fmaf(w, h0.w, a3);
        a4 = fmaf(w, h1.x, a4);
        a5 = fmaf(w, h1.y, a5);
        a6 = fmaf(w, h1.z, a6);
    }
    // wave32 butterfly reduction
    #pragma unroll
    for (int off = 16; off > 0; off >>= 1) {
        a0 += __shfl_xor(a0, off);
        a1 += __shfl_xor(a1, off);
        a2 += __shfl_xor(a2, off);
        a3 += __shfl_xor(a3, off);
        a4 += __shfl_xor(a4, off);
        a5 += __shfl_xor(a5, off);
        a6 += __shfl_xor(a6, off);
    }
    __shared__ float scol[8];
    if (threadIdx.x < 8) scol[threadIdx.x] = 0.f;
    __syncthreads();
    if ((threadIdx.x & 31) == 0) {     // one lane per wave -> ds_add_f32
        atomicAdd(&scol[0], a0);
        atomicAdd(&scol[1], a1);
        atomicAdd(&scol[2], a2);
        atomicAdd(&scol[3], a3);
        atomicAdd(&scol[4], a4);
        atomicAdd(&scol[5], a5);
        atomicAdd(&scol[6], a6);
    }
    __syncthreads();
    if (threadIdx.x < 7) atomicAdd(&colsum[threadIdx.x], scol[threadIdx.x]);
}

// ---------------------------------------------------------------------------
// Kernel 5: out[c] = colsum @ W[:,c] + N * b[c]   (512 outputs)
// ---------------------------------------------------------------------------
__global__ void k_final(const float* __restrict__ colsum, const float* __restrict__ W,
                        const float* __restrict__ b, float* __restrict__ out,
                        int twoH, float fN) {
    int c = blockIdx.x * blockDim.x + threadIdx.x;
    if (c < twoH) {
        float s = fN * b[c];
        #pragma unroll
        for (int j = 0; j < 7; ++j) s = fmaf(colsum[j], W[j * twoH + c], s);
        out[c] = s;
    }
}

// ---------------------------------------------------------------------------
// Launch
// ---------------------------------------------------------------------------
extern "C" void kernel_launch(void* const* d_in, const int* in_sizes, int n_in,
                              void* d_out, int out_size, void* d_ws, size_t ws_size,
                              hipStream_t stream) {
    const float* x   = (const float*)d_in[0];
    const int*   src = (const int*)d_in[1];
    const int*   dst = (const int*)d_in[2];
    const float* Wg  = (const float*)d_in[3];
    const float* bg  = (const float*)d_in[4];
    const float* W   = (const float*)d_in[5];
    const float* b   = (const float*)d_in[6];
    float* out = (float*)d_out;

    const int H = in_sizes[3] / 7;                  // 256
    const long long N = (long long)in_sizes[0] / H; // 100000
    const long long E = in_sizes[1];                // 3200000
    const int twoH = out_size;                      // 512

    // Workspace layout: [colsum 32B][odeg N*4][ideg N*4][hsc N*32 (16B aligned)]
    char* ws = (char*)d_ws;
    float*    colsum = (float*)(ws);
    unsigned* odeg   = (unsigned*)(ws + 32);
    unsigned* ideg   = (unsigned*)(ws + 32 + (size_t)N * 4);
    size_t hs_off = (size_t)(32 + (size_t)N * 8);
    hs_off = (hs_off + 15) & ~(size_t)15;
    float* hsc = (float*)(ws + hs_off);

    // 1) zero colsum + degree counters
    {
        long long words = (long long)(hs_off / 4);
        long long want = (words + 255) / 256;
        int grd = want > 2048 ? 2048 : (int)(want > 0 ? want : 1);
        k_zero<<<grd, 256, 0, stream>>>((unsigned*)ws, words);
    }
    // 2) degrees
    {
        long long want = (E + 255) / 256;
        int grd = want > 2048 ? 2048 : (int)(want > 0 ? want : 1);
        k_degrees<<<grd, 256, 0, stream>>>(src, dst, odeg, ideg, E);
    }
    // 3) gating GEMM (WMMA + async LDS DMA)  (assumes H == 256 per reference)
    {
        int nTiles = (int)((N + 15) / 16);          // 6250
        int grd = nTiles < 1024 ? nTiles : 1024;
        k_gate_wmma<<<grd, 32, 0, stream>>>(x, Wg, bg, odeg, ideg, hsc,
                                            (int)N, nTiles);
    }
    // 4) edge accumulation into colsum
    {
        long long want = (E + 255) / 256;
        int grd = want > 2048 ? 2048 : (int)(want > 0 ? want : 1);
        k_edges<<<grd, 256, 0, stream>>>(src, dst, hsc, colsum, E);
    }
    // 5) finalize 512 outputs
    {
        int grd = (twoH + 255) / 256;
        k_final<<<grd, 256, 0, stream>>>(colsum, W, b, out, twoH, (float)N);
    }
}